// SoftPatternClassifier_28879360098877
// MI455X (gfx1250) — compile-verified
//
#include <hip/hip_runtime.h>
#include <hip/hip_bf16.h>

#define D_DIM 300
#define V_DIM 10000
#define P_DIM 100
#define LM1   6
#define ROWS  (P_DIM * LM1)   /* 600 */
#define B_DIM 32
#define T_DIM 2048
#define NC_DIM 2
#define EPS_F 1e-05f

typedef __attribute__((ext_vector_type(2))) float v2f;
typedef __attribute__((ext_vector_type(8))) float v8f;

// ---------------------------------------------------------------------------
// Kernel 1: transT[v][j] = sum_k diags[j][k] * le[k][v] + bias[j]
// One 16x16 output tile per wave via V_WMMA_F32_16X16X4_F32, K advances by 4.
// A-tile edge masking is done with a 0/1 multiply (no exec-masked loads).
// ---------------------------------------------------------------------------
__global__ __launch_bounds__(128) void gemm_trans_kernel(
    const float* __restrict__ diags,   // (600, 300)
    const float* __restrict__ le,      // (300, 10000)
    const float* __restrict__ bias,    // (600)
    float* __restrict__ transT)        // (10000, 600)
{
  constexpr int Ntiles = V_DIM / 16;        // 625
  constexpr int Mtiles = (ROWS + 15) / 16;  // 38 (rows 600..607 zero-masked)
  const int wave = threadIdx.x >> 5;        // 4 waves / block
  const int lane = threadIdx.x & 31;
  const int tile = blockIdx.x * 4 + wave;   // wave-uniform -> EXEC stays full
  if (tile >= Mtiles * Ntiles) return;
  const int mt = tile / Ntiles, nt = tile % Ntiles;
  const int m0 = mt << 4, n0 = nt << 4;
  const int hi = lane >> 4, lx = lane & 15;

  const int   mrow = m0 + lx;
  const float msk  = (mrow < ROWS) ? 1.0f : 0.0f;   // zero out pad rows
  const float* arow = diags + (size_t)((mrow < ROWS) ? mrow : 0) * D_DIM;
  const int   ncol = n0 + lx;

  v8f acc = {};
  #pragma unroll 15
  for (int k = 0; k < D_DIM; k += 4) {
    const int kk = k + 2 * hi;
    // A 16x4: a[r] = A[m][kk + r] ; unconditional b64 load, then mask.
    const float2 ap = *(const float2*)(arow + kk);
    v2f a = { ap.x * msk, ap.y * msk };
    // B 4x16: b[r] = B[kk + r][n]
    v2f b = { le[(size_t)kk * V_DIM + ncol],
              le[(size_t)(kk + 1) * V_DIM + ncol] };
    acc = __builtin_amdgcn_wmma_f32_16x16x4_f32(
        /*neg_a=*/false, a, /*neg_b=*/false, b,
        /*c_mod=*/(short)0, acc, /*reuse_a=*/false, /*reuse_b=*/false);
  }

  // C/D layout: acc[r] = C[m0 + 8*hi + r][ncol]; store transposed + bias.
  #pragma unroll
  for (int r = 0; r < 8; ++r) {
    const int m = m0 + 8 * hi + r;
    if (m < ROWS) transT[(size_t)ncol * ROWS + m] = acc[r] + bias[m];
  }
}

// ---------------------------------------------------------------------------
// Kernel 2: max-plus scan over T. One thread per (batch, pattern) chain,
// state in registers; branchless software pipeline of the next token's row.
// ---------------------------------------------------------------------------
__global__ __launch_bounds__(128) void scan_kernel(
    const float* __restrict__ transT,  // (10000, 600)
    const float* __restrict__ wild,    // (100, 6)
    const int*   __restrict__ docs,    // (32, 2048)
    const int*   __restrict__ dlens,   // (32)
    float* __restrict__ scores)        // (32, 100)
{
  const int b = blockIdx.x;
  const int p = threadIdx.x;
  if (p >= P_DIM) return;
  const float NINF = -__builtin_inff();
  const int   L    = dlens[b];
  const int   endi = (p < 25) ? 3 : (p < 50) ? 4 : (p < 75) ? 5 : 6;
  const float w0 = wild[p * LM1 + 0], w1 = wild[p * LM1 + 1];
  const float w2 = wild[p * LM1 + 2], w3 = wild[p * LM1 + 3];
  const float w4 = wild[p * LM1 + 4], w5 = wild[p * LM1 + 5];

  float h1 = NINF, h2 = NINF, h3 = NINF, h4 = NINF, h5 = NINF, h6 = NINF;
  float score = NINF;
  const int* drow = docs + (size_t)b * T_DIM;

  const float* rp = transT + (size_t)drow[0] * ROWS + p * LM1;
  float2 c0 = *(const float2*)(rp);
  float2 c1 = *(const float2*)(rp + 2);
  float2 c2 = *(const float2*)(rp + 4);

  for (int t = 0; t < T_DIM - 1; ++t) {
    // next-token row (always in range for t < T-1)
    const float* np = transT + (size_t)drow[t + 1] * ROWS + p * LM1;
    const float2 nx0 = *(const float2*)(np);
    const float2 nx1 = *(const float2*)(np + 2);
    const float2 nx2 = *(const float2*)(np + 4);
    const int tp = (t + 8 < T_DIM) ? (t + 8) : (T_DIM - 1);   // clamped
    __builtin_prefetch(transT + (size_t)drow[tp] * ROWS + p * LM1, 0, 1);

    // hidden[j] = prev[j-1] + max(tm[j-1], wc[j-1]); descending uses old vals
    h6 = h5 + fmaxf(c2.y, w5);
    h5 = h4 + fmaxf(c2.x, w4);
    h4 = h3 + fmaxf(c1.y, w3);
    h3 = h2 + fmaxf(c1.x, w2);
    h2 = h1 + fmaxf(c0.y, w1);
    h1 =      fmaxf(c0.x, w0);   // prev[0] == 0 always
    const float ev = (endi == 3) ? h3 : (endi == 4) ? h4 : (endi == 5) ? h5 : h6;
    if (t < L) score = fmaxf(score, ev);
    c0 = nx0; c1 = nx1; c2 = nx2;
  }
  // epilogue: t = T-1 (no further load)
  {
    const int t = T_DIM - 1;
    h6 = h5 + fmaxf(c2.y, w5);
    h5 = h4 + fmaxf(c2.x, w4);
    h4 = h3 + fmaxf(c1.y, w3);
    h3 = h2 + fmaxf(c1.x, w2);
    h2 = h1 + fmaxf(c0.y, w1);
    h1 =      fmaxf(c0.x, w0);
    const float ev = (endi == 3) ? h3 : (endi == 4) ? h4 : (endi == 5) ? h5 : h6;
    if (t < L) score = fmaxf(score, ev);
  }
  scores[b * P_DIM + p] = score;
}

// ---------------------------------------------------------------------------
// Kernel 3: masked layer-norm over P, threshold at 0, tiny linear head.
// ---------------------------------------------------------------------------
__global__ __launch_bounds__(128) void finalize_kernel(
    const float* __restrict__ scores,   // (32, 100)
    const float* __restrict__ linear_w, // (2, 100)
    const float* __restrict__ linear_b, // (2)
    float* __restrict__ out)            // (32, 2)
{
  __shared__ float red[128];
  const int b = blockIdx.x, tid = threadIdx.x;
  const float INF = __builtin_inff();

  float x = (tid < P_DIM) ? scores[b * P_DIM + tid] : 0.0f;
  const bool  valid = (tid < P_DIM) && (x > -INF) && (x < INF);
  const float xm = valid ? x : 0.0f;

  auto block_sum = [&](float v) -> float {
    red[tid] = v;
    __syncthreads();
    for (int s = 64; s > 0; s >>= 1) {
      if (tid < s) red[tid] += red[tid + s];
      __syncthreads();
    }
    const float r = red[0];
    __syncthreads();
    return r;
  };

  const float cnt  = block_sum(valid ? 1.0f : 0.0f);
  const float mean = block_sum(xm) / cnt;
  const float dv   = valid ? (xm - mean) * (xm - mean) : 0.0f;
  const float var  = block_sum(dv) / cnt;
  const float nrm  = (xm - mean) * rsqrtf(var + EPS_F);
  const float res  = valid ? nrm : x;        // keep +/-inf where masked
  const float ind  = (res > 0.0f) ? 1.0f : 0.0f;

  for (int c = 0; c < NC_DIM; ++c) {
    const float v = (tid < P_DIM) ? ind * linear_w[c * P_DIM + tid] : 0.0f;
    const float s = block_sum(v);
    if (tid == 0) out[b * NC_DIM + c] = s + linear_b[c];
  }
}

// ---------------------------------------------------------------------------
extern "C" void kernel_launch(void* const* d_in, const int* in_sizes, int n_in,
                              void* d_out, int out_size, void* d_ws, size_t ws_size,
                              hipStream_t stream)
{
  const float* le    = (const float*)d_in[0];  // local_embeddings (300,10000)
  const float* diags = (const float*)d_in[1];  // (600,300)
  const float* bias  = (const float*)d_in[2];  // (600,1)
  const float* wild  = (const float*)d_in[3];  // (100,6)
  const float* lw    = (const float*)d_in[4];  // (2,100)
  const float* lb    = (const float*)d_in[5];  // (2)
  const int*   docs  = (const int*)d_in[6];    // (32,2048)
  const int*   dlens = (const int*)d_in[7];    // (32)
  float* out = (float*)d_out;

  float* transT = (float*)d_ws;                     // 10000*600 f32 = 24 MB
  float* scores = transT + (size_t)V_DIM * ROWS;    // 32*100 f32

  constexpr int tiles = ((ROWS + 15) / 16) * (V_DIM / 16);  // 38*625
  hipLaunchKernelGGL(gemm_trans_kernel, dim3((tiles + 3) / 4), dim3(128), 0,
                     stream, diags, le, bias, transT);
  hipLaunchKernelGGL(scan_kernel, dim3(B_DIM), dim3(128), 0, stream,
                     transT, wild, docs, dlens, scores);
  hipLaunchKernelGGL(finalize_kernel, dim3(B_DIM), dim3(128), 0, stream,
                     scores, lw, lb, out);
  (void)in_sizes; (void)n_in; (void)out_size; (void)ws_size;
}